// YOLOLoss_22153441313087
// MI455X (gfx1250) — compile-verified
//
#include <hip/hip_runtime.h>

typedef __attribute__((ext_vector_type(2))) float v2f;
typedef __attribute__((ext_vector_type(8))) float v8f;
typedef int b128_t __attribute__((vector_size(16)));   // matches builtin's V4i param

#define NUM_CLASSES 80
#define BBOX_ATTRS  85
#define BS      32
#define NA      3
#define HW      76
#define PLANE   (HW*HW)                    // 5776
#define NGT     50
#define GTF     8                          // floats per GT record
#define BLOCK   256
#define BPP     ((PLANE + BLOCK - 1)/BLOCK) // 23 blocks per (b,a) plane
#define NPLANES (BS*NA)                     // 96
#define NBLK    (BPP*NPLANES)               // 2208 partial records
#define NPAD    2240                        // padded stride (multiple of 64)
#define NACC    5
#define EPSV    1e-7f

__device__ __constant__ float d_anc_w[9] = {12.f,19.f,40.f,36.f,76.f,72.f,142.f,192.f,459.f};
__device__ __constant__ float d_anc_h[9] = {16.f,36.f,28.f,75.f,55.f,146.f,110.f,243.f,401.f};
// scaled anchors for l=2 (stride 608/76 = 8): ANCHORS[0..2] / 8
__device__ __constant__ float d_aw[3] = {1.5f, 2.375f, 5.0f};
__device__ __constant__ float d_ah[3] = {2.0f, 4.5f, 3.5f};

__device__ __forceinline__ float sigmoidf_(float x){ return 1.0f/(1.0f + __expf(-x)); }
__device__ __forceinline__ float bce_(float p, float t){
  p = fminf(fmaxf(p, EPSV), 1.0f - EPSV);
  return -(t*__logf(p) + (1.0f - t)*__logf(1.0f - p));
}

// ---------------------------------------------------------------------------
// Kernel 0: per-GT anchor assignment. gtinfo[b][g] = {cx,cy,bw,bh,cls,kf,i,j}
// ---------------------------------------------------------------------------
__global__ void yolo_assign_kernel(const float* __restrict__ targets,
                                   float* __restrict__ gtinfo) {
  int t = blockIdx.x * blockDim.x + threadIdx.x;
  if (t >= BS * NGT) return;
  const float* tp = targets + (size_t)t * 5;
  float cx = tp[0] * (float)HW;
  float cy = tp[1] * (float)HW;
  float bw = tp[2] * (float)HW;
  float bh = tp[3] * (float)HW;
  float cls = tp[4];
  // IoU of wh-box (grid units) vs raw-pixel anchors, centered at origin
  // (reference uses raw anchors here; replicate exactly)
  float best = -1.0f; int bn = 0;
  float garea = bw * bh;
#pragma unroll
  for (int a = 0; a < 9; a++) {
    float aw = d_anc_w[a], ah = d_anc_h[a];
    float inter = fminf(bw, aw) * fminf(bh, ah);
    float iou = inter / (garea + aw * ah - inter);
    if (iou > best) { best = iou; bn = a; }   // strict '>' => first max (argmax)
  }
  float kf = (bn < 3) ? (float)bn : -1.0f;     // ANCHORS_MASK[2] = [0,1,2]
  int i = min(max((int)floorf(cx), 0), HW - 1);
  int j = min(max((int)floorf(cy), 0), HW - 1);
  float* o = gtinfo + (size_t)t * GTF;
  o[0] = cx; o[1] = cy; o[2] = bw; o[3] = bh;
  o[4] = cls; o[5] = kf; o[6] = (float)i; o[7] = (float)j;
}

// ---------------------------------------------------------------------------
// Kernel 1: per-cell pass. grid = (BPP, BS*NA). Reads ch 0..4 everywhere,
// ch 5..84 only at assigned (obj) cells. Block-reduces 5 accumulators.
// GT table staged into LDS via GLOBAL_LOAD_ASYNC_TO_LDS (ASYNCcnt-tracked
// memory->LDS DMA) when the toolchain exposes the gfx1250 builtins.
// ---------------------------------------------------------------------------
__global__ void yolo_main_kernel(const float* __restrict__ inp,
                                 const float* __restrict__ gtinfo,
                                 float* __restrict__ partial) {
  __shared__ float gsh[NGT * GTF];
  __shared__ float red[NACC][BLOCK];

  const int tid   = threadIdx.x;
  const int plane = blockIdx.y;            // 0..95
  const int b     = plane / NA;
  const int a     = plane % NA;
  const int p     = blockIdx.x * BLOCK + tid;
  const bool act  = (p < PLANE);

  // ---- stage this batch's GT table (1600 B) into LDS -----------------------
#if __has_builtin(__builtin_amdgcn_global_load_async_to_lds_b128)
  if (tid < (NGT * GTF) / 4) {             // 100 x 16B async DMA transfers
    const float* src = gtinfo + (size_t)b * NGT * GTF + tid * 4;
    __builtin_amdgcn_global_load_async_to_lds_b128(
        (__attribute__((address_space(1))) b128_t*)src,
        (__attribute__((address_space(3))) b128_t*)&gsh[tid * 4],
        /*offset=*/0, /*cpol=*/0);
  }
#if __has_builtin(__builtin_amdgcn_s_wait_asynccnt)
  __builtin_amdgcn_s_wait_asynccnt(0);
#else
  asm volatile("s_wait_asynccnt 0" ::: "memory");
#endif
  __syncthreads();
#else
  for (int s = tid; s < NGT * GTF; s += BLOCK)
    gsh[s] = gtinfo[(size_t)b * NGT * GTF + s];
  __syncthreads();
#endif

  float accN = 0.f, accLoc = 0.f, accCls = 0.f, accConf = 0.f, accSel = 0.f;

  if (act) {
    const int j = p / HW, i = p % HW;
    const float* base =
        inp + ((size_t)b * NA * BBOX_ATTRS + (size_t)a * BBOX_ATTRS) * PLANE + p;
    __builtin_prefetch(base, 0, 0);
    __builtin_prefetch(base + 5 * PLANE, 0, 0);   // class channels (sparse use)

    float p0 = base[0 * PLANE];
    float p1 = base[1 * PLANE];
    float p2 = base[2 * PLANE];
    float p3 = base[3 * PLANE];
    float p4 = base[4 * PLANE];

    float x = sigmoidf_(p0), y = sigmoidf_(p1);
    float conf = sigmoidf_(p4);
    float px = x + (float)i, py = y + (float)j;
    float pw = __expf(p2) * d_aw[a];
    float ph = __expf(p3) * d_ah[a];
    float pxmin = px - 0.5f * pw, pxmax = px + 0.5f * pw;
    float pymin = py - 0.5f * ph, pymax = py + 0.5f * ph;
    float parea = pw * ph;

    float ioumax = 0.0f;          // all IoUs >= 0
    bool assigned = false;
    float tcx = 0.f, tcy = 0.f, tbw = 0.f, tbh = 0.f;
    int tcls = 0;
    float fa = (float)a, fi = (float)i, fj = (float)j;

    for (int t = 0; t < NGT; t++) {
      const float* g = &gsh[t * GTF];
      float gcx = g[0], gcy = g[1], gbw = g[2], gbh = g[3];
      float gxmin = gcx - 0.5f * gbw, gxmax = gcx + 0.5f * gbw;
      float gymin = gcy - 0.5f * gbh, gymax = gcy + 0.5f * gbh;
      float iw = fmaxf(fminf(pxmax, gxmax) - fmaxf(pxmin, gxmin), 0.f);
      float ih = fmaxf(fminf(pymax, gymax) - fmaxf(pymin, gymin), 0.f);
      float inter = iw * ih;
      float iou = inter / (parea + gbw * gbh - inter);   // iou_xywh: minus
      ioumax = fmaxf(ioumax, iou);
      if (g[5] == fa && g[6] == fi && g[7] == fj) {       // scatter: last wins
        assigned = true;
        tcx = gcx; tcy = gcy; tbw = gbw; tbh = gbh; tcls = (int)g[4];
      }
    }

    float objf = assigned ? 1.f : 0.f;
    float sel  = (assigned || ioumax <= 0.5f) ? 1.f : 0.f;  // noobj || obj
    accN = objf;
    accSel = sel;
    accConf = bce_(conf, objf) * sel;

    if (assigned) {
      // box_ciou (replicates reference, incl. union = a1 + a2 + inter)
      float txmin = tcx - 0.5f * tbw, txmax = tcx + 0.5f * tbw;
      float tymin = tcy - 0.5f * tbh, tymax = tcy + 0.5f * tbh;
      float iw = fmaxf(fminf(pxmax, txmax) - fmaxf(pxmin, txmin), 0.f);
      float ih = fmaxf(fminf(pymax, tymax) - fmaxf(pymin, tymin), 0.f);
      float inter = iw * ih;
      float uni = parea + tbw * tbh + inter;              // reference's '+'
      float iou = inter / uni;
      float dx = px - tcx, dy = py - tcy;
      float cdist = dx * dx + dy * dy;
      float ew = fmaxf(fmaxf(pxmax, txmax) - fminf(pxmin, txmin), 0.f);
      float eh = fmaxf(fmaxf(pymax, tymax) - fminf(pymin, tymin), 0.f);
      float diag = ew * ew + eh * eh;
      float ciou = iou - cdist / fmaxf(diag, 1e-6f);
      float da = atanf(pw / fmaxf(ph, 1e-6f)) - atanf(tbw / fmaxf(tbh, 1e-6f));
      float v = 0.40528473456935109f * da * da;           // 4/pi^2
      float alpha = v / fmaxf(1.f - iou + v, 1e-6f);
      ciou -= alpha * v;
      accLoc = 1.f - ciou;

      // class BCE: only obj cells touch channels 5..84
      float s = 0.f;
      for (int c = 0; c < NUM_CLASSES; c++) {
        float pc = sigmoidf_(base[(size_t)(5 + c) * PLANE]);
        float tt = (c == tcls) ? 1.f : 0.f;
        s += bce_(pc, tt);
      }
      accCls = s;
    }
  }

  // block tree-reduce 5 accumulators
  red[0][tid] = accN; red[1][tid] = accLoc; red[2][tid] = accCls;
  red[3][tid] = accConf; red[4][tid] = accSel;
  __syncthreads();
  for (int st = BLOCK / 2; st > 0; st >>= 1) {
    if (tid < st) {
#pragma unroll
      for (int k = 0; k < NACC; k++) red[k][tid] += red[k][tid + st];
    }
    __syncthreads();
  }
  if (tid < NACC) {
    int blk = blockIdx.y * gridDim.x + blockIdx.x;   // 0..NBLK-1
    partial[(size_t)tid * NPAD + blk] = red[tid][0];
  }
}

// ---------------------------------------------------------------------------
// Kernel 2: final reduction on the matrix unit. Each wave w < NACC reduces
// one accumulator column with V_WMMA_F32_16X16X4_F32 (B = ones): 64 fp32
// summed per issue, accumulated in C; cross-lane finish via v_readlane.
// ---------------------------------------------------------------------------
__global__ void yolo_final_kernel(const float* __restrict__ partial,
                                  float* __restrict__ out) {
  __shared__ float finals[NACC];
  const int tid  = threadIdx.x;
  const int w    = tid >> 5;
  const int lane = tid & 31;

  if (w < NACC) {
    const float* col = partial + (size_t)w * NPAD;
    v8f c = {0.f, 0.f, 0.f, 0.f, 0.f, 0.f, 0.f, 0.f};
    v2f bones = {1.f, 1.f};
    for (int base = 0; base < NPAD; base += 64) {
      int i0 = base + lane, i1 = base + 32 + lane;
      v2f av;
      av.x = (i0 < NBLK) ? col[i0] : 0.f;
      av.y = (i1 < NBLK) ? col[i1] : 0.f;
      // D = A(16x4) * ones(4x16) + C : every A element lands in exactly one
      // row-sum; C accumulates across chunks.
      c = __builtin_amdgcn_wmma_f32_16x16x4_f32(
          /*neg_a=*/false, av, /*neg_b=*/false, bones,
          /*c_mod=*/(short)0, c, /*reuse_a=*/false, /*reuse_b=*/false);
    }
    // lane n (0..15): sum of rows 0..7 at col n; lane 16+n: rows 8..15.
    float s = c[0] + c[1] + c[2] + c[3] + c[4] + c[5] + c[6] + c[7];
    float t0  = __int_as_float(__builtin_amdgcn_readlane(__float_as_int(s), 0));
    float t16 = __int_as_float(__builtin_amdgcn_readlane(__float_as_int(s), 16));
    if (lane == 0) finals[w] = t0 + t16;
  }
  __syncthreads();

  if (tid == 0) {
    float n     = finals[0];
    float sloc  = finals[1];
    float scls  = finals[2];
    float sconf = finals[3];
    float ssel  = finals[4];
    float nf = fmaxf(n, 1.f);
    float loss_loc = sloc / nf;
    float loss_cls = scls / (nf * (float)NUM_CLASSES);
    float has = (n > 0.f) ? 1.f : 0.f;
    const float BOX_RATIO = 0.05f;
    const float CLS_RATIO = 1.0f;                           // 80/80
    const float OBJ_RATIO = 5.0f * (608.f * 608.f) / (416.f * 416.f);
    const float BAL2 = 4.0f;                                // BALANCE[2]
    float loss_conf = sconf / fmaxf(ssel, 1.f);
    out[0] = has * (loss_loc * BOX_RATIO + loss_cls * CLS_RATIO)
           + loss_conf * BAL2 * OBJ_RATIO;
  }
}

// ---------------------------------------------------------------------------
extern "C" void kernel_launch(void* const* d_in, const int* in_sizes, int n_in,
                              void* d_out, int out_size, void* d_ws, size_t ws_size,
                              hipStream_t stream) {
  // d_in[0]: l (int scalar, == 2 per setup; constants baked for l=2)
  const float* inp     = (const float*)d_in[1];   // (32, 255, 76, 76) f32
  const float* targets = (const float*)d_in[2];   // (32, 50, 5) f32
  (void)in_sizes; (void)n_in; (void)out_size; (void)ws_size;

  float* gtinfo  = (float*)d_ws;                  // 1600 * 8 f32  = 51.2 KB
  float* partial = gtinfo + BS * NGT * GTF;       // 5 * 2240 f32  = 44.8 KB

  yolo_assign_kernel<<<(BS * NGT + 255) / 256, 256, 0, stream>>>(targets, gtinfo);

  dim3 grid(BPP, NPLANES);
  yolo_main_kernel<<<grid, BLOCK, 0, stream>>>(inp, gtinfo, partial);

  yolo_final_kernel<<<1, 256, 0, stream>>>(partial, (float*)d_out);
}